// TwoDPositionalEncoding_74775380624072
// MI455X (gfx1250) — compile-verified
//
#include <hip/hip_runtime.h>
#include <stdint.h>

// 2D positional encoding: out[row,:] = x_enc[32+xi(row), :] + y_enc[32+yi(row), :]
// Pure store-bandwidth bound (256 MiB output @ 23.3 TB/s ~ 11 us).
// CDNA5 path: async global->LDS staging of the 18 KiB hot table rows,
// wave32 row ownership with scalarized index math, b128 LDS reads,
// non-temporal b128 global stores.

#define ROWS      (64 * 4096)
#define D_MODEL   256
#define TAB_ROWS  9          // xi,yi in [0,8]
#define DELTA     32
#define BLOCK     256
#define GRID      4096

typedef __attribute__((ext_vector_type(4))) float v4f;
typedef __attribute__((ext_vector_type(4))) int   v4i;

#define AS1 __attribute__((address_space(1)))
#define AS3 __attribute__((address_space(3)))

__global__ __launch_bounds__(BLOCK) void pos_enc_kernel(
    const float* __restrict__ tokens,   // [ROWS, 2]
    const float* __restrict__ xe,       // [64, 256]
    const float* __restrict__ ye,       // [64, 256]
    float* __restrict__ out)            // [ROWS, 256]
{
    __shared__ __align__(16) float ldsX[TAB_ROWS * D_MODEL];  // 9 KiB
    __shared__ __align__(16) float ldsY[TAB_ROWS * D_MODEL];  // 9 KiB

    const int tid = threadIdx.x;
    const float* xsrc = xe + DELTA * D_MODEL;  // rows 32..40 only
    const float* ysrc = ye + DELTA * D_MODEL;

    // ---- Stage hot table rows into LDS (async tensor path when available) ----
#if __has_builtin(__builtin_amdgcn_global_load_async_to_lds_b128)
    for (int i = tid; i < (TAB_ROWS * D_MODEL) / 4; i += BLOCK) {
        __builtin_amdgcn_global_load_async_to_lds_b128(
            (AS1 v4i*)(xsrc + 4 * i), (AS3 v4i*)(ldsX + 4 * i), 0, 0);
        __builtin_amdgcn_global_load_async_to_lds_b128(
            (AS1 v4i*)(ysrc + 4 * i), (AS3 v4i*)(ldsY + 4 * i), 0, 0);
    }
  #if __has_builtin(__builtin_amdgcn_s_wait_asynccnt)
    __builtin_amdgcn_s_wait_asynccnt(0);
  #else
    asm volatile("s_wait_asynccnt 0" ::: "memory");
  #endif
#else
    for (int i = tid; i < (TAB_ROWS * D_MODEL) / 4; i += BLOCK) {
        ((v4f*)ldsX)[i] = ((const v4f*)xsrc)[i];
        ((v4f*)ldsY)[i] = ((const v4f*)ysrc)[i];
    }
#endif
    __syncthreads();

    // ---- One wave32 per output row; lane L owns bytes [32L, 32L+32) ----
    const int lane = tid & 31;
    const int wid  = __builtin_amdgcn_readfirstlane(tid >> 5);  // force SGPR
    const int rowStride = GRID * (BLOCK / 32);

    for (int row = blockIdx.x * (BLOCK / 32) + wid; row < ROWS; row += rowStride) {
        // Wave-uniform token fetch -> scalar load + scalar index math.
        const float tx = tokens[2 * row + 0];
        const float ty = tokens[2 * row + 1];
        int xi = (int)(tx * 9.0f);  if (xi > 8) xi = 8;   // trunc matches ref
        int yi = (int)(ty * 9.0f);  if (yi > 8) yi = 8;

        const v4f* xr = (const v4f*)(ldsX + xi * D_MODEL);
        const v4f* yr = (const v4f*)(ldsY + yi * D_MODEL);

        // 2x ds_load_b128 per table, 8 f32 adds, 2x non-temporal b128 stores.
        const v4f a0 = xr[2 * lane];
        const v4f a1 = xr[2 * lane + 1];
        const v4f b0 = yr[2 * lane];
        const v4f b1 = yr[2 * lane + 1];
        const v4f r0 = a0 + b0;
        const v4f r1 = a1 + b1;

        v4f* orow = (v4f*)(out + (size_t)row * D_MODEL);
        __builtin_nontemporal_store(r0, orow + 2 * lane);
        __builtin_nontemporal_store(r1, orow + 2 * lane + 1);
    }
}

extern "C" void kernel_launch(void* const* d_in, const int* in_sizes, int n_in,
                              void* d_out, int out_size, void* d_ws, size_t ws_size,
                              hipStream_t stream) {
    const float* tokens = (const float*)d_in[0];  // [64,4096,2] f32
    const float* xe     = (const float*)d_in[1];  // [64,256]    f32
    const float* ye     = (const float*)d_in[2];  // [64,256]    f32
    float*       out    = (float*)d_out;          // [64,4096,256] f32

    hipLaunchKernelGGL(pos_enc_kernel, dim3(GRID), dim3(BLOCK), 0, stream,
                       tokens, xe, ye, out);
}